// SimpleGNN_54090818126571
// MI455X (gfx1250) — compile-verified
//
#include <hip/hip_runtime.h>

typedef __attribute__((ext_vector_type(2))) float v2f;
typedef __attribute__((ext_vector_type(8))) float v8f;

// ---------------------------------------------------------------------------
// Elementwise / degree kernels
// ---------------------------------------------------------------------------
__global__ void k_zero2(float* __restrict__ a, float* __restrict__ b, int n) {
    int i = blockIdx.x * blockDim.x + threadIdx.x;
    if (i < n) { a[i] = 0.0f; b[i] = 0.0f; }
}

__global__ void k_degrees(const int* __restrict__ src, const int* __restrict__ dst,
                          float* __restrict__ dout, float* __restrict__ din, int E) {
    int e = blockIdx.x * blockDim.x + threadIdx.x;
    if (e < E) {
        atomicAdd(&dout[src[e]], 1.0f);
        atomicAdd(&din[dst[e]], 1.0f);
    }
}

__global__ void k_rsqrt2(float* __restrict__ a, float* __restrict__ b, int n) {
    int i = blockIdx.x * blockDim.x + threadIdx.x;
    if (i < n) {
        float x = a[i]; a[i] = rsqrtf(x < 1.0f ? 1.0f : x);
        float y = b[i]; b[i] = rsqrtf(y < 1.0f ? 1.0f : y);
    }
}

// out[i] = bias[i & mask]  (mask = ncols-1, ncols power of two)
__global__ void k_bias_init(float* __restrict__ out, const float* __restrict__ bias,
                            int mask, int total) {
    int i = blockIdx.x * blockDim.x + threadIdx.x;
    if (i < total) out[i] = bias[i & mask];
}

// ---------------------------------------------------------------------------
// GEMM1: H[r,0:64] = (X[r,0:64] @ W[64,64]) * rs_out[r]
// One wave -> 16 output rows; 4 column tiles of 16; K=64 in 16 steps of 4.
// W staged in LDS with K-pairs interleaved: sWt[p*128 + c*2 + j] = W[2p+j][c]
// so each B fragment is a single aligned ds_load_b64 into an even VGPR pair.
// ---------------------------------------------------------------------------
__global__ __launch_bounds__(256) void k_gemm1(
    const float* __restrict__ X, const float* __restrict__ W,
    const float* __restrict__ rs_out, float* __restrict__ H, int N) {
    __shared__ float sWt[64 * 64];
    const int tid = threadIdx.x;
#pragma unroll
    for (int i = 0; i < 16; ++i) {       // 4096 elems / 256 threads
        int idx = tid + 256 * i;         // coalesced global read
        int r = idx >> 6, c = idx & 63;
        sWt[(r >> 1) * 128 + c * 2 + (r & 1)] = W[idx];
    }
    __syncthreads();

    const int wave = tid >> 5;
    const int lane = tid & 31;
    const int lo = lane & 15;      // column / M-low index
    const int hi = lane >> 4;      // K-slot group
    const int r0 = (blockIdx.x * 8 + wave) * 16;
    if (r0 >= N) return;           // wave-uniform: EXEC stays all-ones below

    int arow = r0 + lo; if (arow >= N) arow = N - 1;   // clamp (rows not stored)
    const float* xrow = X + (size_t)arow * 64 + 2 * hi;

    v2f a[16];
#pragma unroll
    for (int ks = 0; ks < 16; ++ks)
        a[ks] = *(const v2f*)(xrow + 4 * ks);

    v8f c0 = {}, c1 = {}, c2 = {}, c3 = {};
#pragma unroll
    for (int ks = 0; ks < 16; ++ks) {
        const float* w0 = &sWt[(2 * ks + hi) * 128 + lo * 2];
        v2f b0 = *(const v2f*)(w0 + 0);
        v2f b1 = *(const v2f*)(w0 + 32);
        v2f b2 = *(const v2f*)(w0 + 64);
        v2f b3 = *(const v2f*)(w0 + 96);
        c0 = __builtin_amdgcn_wmma_f32_16x16x4_f32(false, a[ks], false, b0, (short)0, c0, false, false);
        c1 = __builtin_amdgcn_wmma_f32_16x16x4_f32(false, a[ks], false, b1, (short)0, c1, false, false);
        c2 = __builtin_amdgcn_wmma_f32_16x16x4_f32(false, a[ks], false, b2, (short)0, c2, false, false);
        c3 = __builtin_amdgcn_wmma_f32_16x16x4_f32(false, a[ks], false, b3, (short)0, c3, false, false);
    }

#pragma unroll
    for (int v = 0; v < 8; ++v) {
        const int row = r0 + v + 8 * hi;
        if (row < N) {
            const float s = rs_out[row];
            float* hrow = H + (size_t)row * 64 + lo;
            hrow[0]  = c0[v] * s;
            hrow[16] = c1[v] * s;
            hrow[32] = c2[v] * s;
            hrow[48] = c3[v] * s;
        }
    }
}

// ---------------------------------------------------------------------------
// GEMM2: H2[r,0:32] = (relu(A[r,0:64]) @ W[64,32]) * rs_out[r]
// Same K-pair-interleaved LDS layout: sWt[p*64 + c*2 + j] = W[2p+j][c]
// ---------------------------------------------------------------------------
__global__ __launch_bounds__(256) void k_gemm2(
    const float* __restrict__ A, const float* __restrict__ W,
    const float* __restrict__ rs_out, float* __restrict__ H2, int N) {
    __shared__ float sWt[64 * 32];
    const int tid = threadIdx.x;
#pragma unroll
    for (int i = 0; i < 8; ++i) {        // 2048 elems / 256 threads
        int idx = tid + 256 * i;
        int r = idx >> 5, c = idx & 31;
        sWt[(r >> 1) * 64 + c * 2 + (r & 1)] = W[idx];
    }
    __syncthreads();

    const int wave = tid >> 5;
    const int lane = tid & 31;
    const int lo = lane & 15;
    const int hi = lane >> 4;
    const int r0 = (blockIdx.x * 8 + wave) * 16;
    if (r0 >= N) return;

    int arow = r0 + lo; if (arow >= N) arow = N - 1;
    const float* xrow = A + (size_t)arow * 64 + 2 * hi;

    v2f a[16];
#pragma unroll
    for (int ks = 0; ks < 16; ++ks) {
        v2f t = *(const v2f*)(xrow + 4 * ks);
        t.x = fmaxf(t.x, 0.0f);            // fused ReLU
        t.y = fmaxf(t.y, 0.0f);
        a[ks] = t;
    }

    v8f c0 = {}, c1 = {};
#pragma unroll
    for (int ks = 0; ks < 16; ++ks) {
        const float* w0 = &sWt[(2 * ks + hi) * 64 + lo * 2];
        v2f b0 = *(const v2f*)(w0 + 0);
        v2f b1 = *(const v2f*)(w0 + 32);
        c0 = __builtin_amdgcn_wmma_f32_16x16x4_f32(false, a[ks], false, b0, (short)0, c0, false, false);
        c1 = __builtin_amdgcn_wmma_f32_16x16x4_f32(false, a[ks], false, b1, (short)0, c1, false, false);
    }

#pragma unroll
    for (int v = 0; v < 8; ++v) {
        const int row = r0 + v + 8 * hi;
        if (row < N) {
            const float s = rs_out[row];
            float* hrow = H2 + (size_t)row * 32 + lo;
            hrow[0]  = c0[v] * s;
            hrow[16] = c1[v] * s;
        }
    }
}

// ---------------------------------------------------------------------------
// Edge scatter-add: agg[dst] += H[src] * rs_in[dst]   (agg pre-seeded w/ bias)
// 16 lanes x float4 per edge (64 feats) / 8 lanes x float4 (32 feats)
// ---------------------------------------------------------------------------
__global__ void k_scatter64(const int* __restrict__ src, const int* __restrict__ dst,
                            const float* __restrict__ H, const float* __restrict__ rs_in,
                            float* __restrict__ agg, int E) {
    int t = blockIdx.x * blockDim.x + threadIdx.x;
    int e = t >> 4;
    if (e >= E) return;
    int q = (t & 15) * 4;
    int s = src[e], d = dst[e];
    float r = rs_in[d];
    float4 h = *(const float4*)(H + (size_t)s * 64 + q);
    float* o = agg + (size_t)d * 64 + q;
    atomicAdd(o + 0, h.x * r);
    atomicAdd(o + 1, h.y * r);
    atomicAdd(o + 2, h.z * r);
    atomicAdd(o + 3, h.w * r);
}

__global__ void k_scatter32(const int* __restrict__ src, const int* __restrict__ dst,
                            const float* __restrict__ H, const float* __restrict__ rs_in,
                            float* __restrict__ agg, int E) {
    int t = blockIdx.x * blockDim.x + threadIdx.x;
    int e = t >> 3;
    if (e >= E) return;
    int q = (t & 7) * 4;
    int s = src[e], d = dst[e];
    float r = rs_in[d];
    float4 h = *(const float4*)(H + (size_t)s * 32 + q);
    float* o = agg + (size_t)d * 32 + q;
    atomicAdd(o + 0, h.x * r);
    atomicAdd(o + 1, h.y * r);
    atomicAdd(o + 2, h.z * r);
    atomicAdd(o + 3, h.w * r);
}

// ---------------------------------------------------------------------------
extern "C" void kernel_launch(void* const* d_in, const int* in_sizes, int n_in,
                              void* d_out, int out_size, void* d_ws, size_t ws_size,
                              hipStream_t stream) {
    (void)n_in; (void)out_size; (void)ws_size;
    const float* x  = (const float*)d_in[0];
    const int*   ei = (const int*)d_in[1];   // [2, E] int32 (JAX default x64-off)
    const float* W1 = (const float*)d_in[2];
    const float* b1 = (const float*)d_in[3];
    const float* W2 = (const float*)d_in[4];
    const float* b2 = (const float*)d_in[5];
    float* out = (float*)d_out;

    const int N = in_sizes[0] / 64;
    const int E = in_sizes[1] / 2;
    const int* src = ei;
    const int* dst = ei + E;

    float* ws = (float*)d_ws;
    float* rsdout = ws;                    // N   (deg_out accum -> rsqrt)
    float* rsdin  = ws + (size_t)1 * N;    // N   (deg_in  accum -> rsqrt)
    float* h1s    = ws + (size_t)2 * N;    // 64N (scaled layer-1 GEMM out)
    float* agg1   = ws + (size_t)66 * N;   // 64N (layer-1 aggregation + b1)
    float* h2s    = ws + (size_t)130 * N;  // 32N (scaled layer-2 GEMM out)

    const dim3 B(256);
    const int ntiles = (N + 15) / 16;
    const int gemmBlocks = (ntiles + 7) / 8;

    k_zero2   <<<(N + 255) / 256, B, 0, stream>>>(rsdout, rsdin, N);
    k_degrees <<<(E + 255) / 256, B, 0, stream>>>(src, dst, rsdout, rsdin, E);
    k_rsqrt2  <<<(N + 255) / 256, B, 0, stream>>>(rsdout, rsdin, N);

    k_bias_init<<<(N * 64 + 255) / 256, B, 0, stream>>>(agg1, b1, 63, N * 64);
    k_bias_init<<<(N * 32 + 255) / 256, B, 0, stream>>>(out,  b2, 31, N * 32);

    k_gemm1<<<gemmBlocks, B, 0, stream>>>(x, W1, rsdout, h1s, N);
    k_scatter64<<<(int)(((long long)E * 16 + 255) / 256), B, 0, stream>>>(src, dst, h1s, rsdin, agg1, E);

    k_gemm2<<<gemmBlocks, B, 0, stream>>>(agg1, W2, rsdout, h2s, N);
    k_scatter32<<<(int)(((long long)E * 8 + 255) / 256), B, 0, stream>>>(src, dst, h2s, rsdin, out, E);
}